// LightGCN_79147657330993
// MI455X (gfx1250) — compile-verified
//
#include <hip/hip_runtime.h>
#include <hip/hip_bf16.h>

#define N_NODES 100000
#define N_EDGES 1600000
#define IN_F 128
#define H_F 64

typedef float v2f __attribute__((ext_vector_type(2)));
typedef float v8f __attribute__((ext_vector_type(8)));

// ---------------- degree / normalization ----------------

__global__ void k_zero_i32(int* __restrict__ p, int n) {
    int i = blockIdx.x * blockDim.x + threadIdx.x;
    if (i < n) p[i] = 0;
}

__global__ void k_zero_f4(float4* __restrict__ p, int n4) {
    int i = blockIdx.x * blockDim.x + threadIdx.x;
    if (i < n4) p[i] = make_float4(0.f, 0.f, 0.f, 0.f);
}

__global__ void k_count_deg(const int* __restrict__ dst, int* __restrict__ cnt, int nE) {
    int e = blockIdx.x * blockDim.x + threadIdx.x;
    if (e < nE) atomicAdd(&cnt[dst[e]], 1);
}

__global__ void k_dinv(const int* __restrict__ cnt, float* __restrict__ dinv1,
                       float* __restrict__ dinv2, int n) {
    int v = blockIdx.x * blockDim.x + threadIdx.x;
    if (v >= n) return;
    int c = cnt[v];
    dinv1[v] = rsqrtf((float)(c + 1));            // with self-loop: deg >= 1 always
    dinv2[v] = (c > 0) ? rsqrtf((float)c) : 0.f;  // no self-loop
}

// ---------------- WMMA GEMM: X[N,64] = A[N,128] * W[128,64] (f32) ----------------
// One wave -> 16 rows x 64 cols (4 accumulator tiles). Block = 256 thr = 8 waves -> 128 rows.
// LDS layout for W is K-pair interleaved: element (k, n) lives at
//   Wl[(k/2)*128 + 2*n + (k&1)]
// so a B fragment {W[k0][n], W[k0+1][n]} is one contiguous 64-bit LDS load.

__global__ __launch_bounds__(256) void k_gemm_wmma(const float* __restrict__ A,
                                                   const float* __restrict__ W,
                                                   float* __restrict__ X, int nRows) {
    __shared__ float Wl[IN_F * H_F];  // 32 KB of the 320 KB/WGP LDS
    for (int idx = threadIdx.x; idx < IN_F * H_F; idx += 256) {
        int k = idx >> 6;        // row of W  (K index)
        int n = idx & 63;        // col of W  (N index)
        Wl[((k >> 1) << 7) + (n << 1) + (k & 1)] = W[idx];
    }
    __syncthreads();

    const int lane = threadIdx.x & 31;
    const int wave = threadIdx.x >> 5;
    const int half = lane >> 4;   // 0: lanes 0-15 (K pair 0/1), 1: lanes 16-31 (K pair 2/3)
    const int m    = lane & 15;

    const int rowBase = blockIdx.x * 128 + wave * 16;
    const int rowLd   = min(rowBase + m, nRows - 1);  // clamp loads; stores guarded
    const float* arow = A + (size_t)rowLd * IN_F;

    v8f acc0 = {}, acc1 = {}, acc2 = {}, acc3 = {};

    #pragma unroll 4
    for (int kb = 0; kb < IN_F; kb += 4) {
        const int k0 = kb + half * 2;
        // A fragment (16x4 f32): one b64 global load per lane
        v2f a = *(const v2f*)(arow + k0);
        // B fragments: one b64 LDS load each (K-pair interleaved layout)
        const float* wp = &Wl[(k0 >> 1) << 7];   // (k0/2)*128
        v2f b0 = *(const v2f*)(wp + (m << 1));
        v2f b1 = *(const v2f*)(wp + ((m + 16) << 1));
        v2f b2 = *(const v2f*)(wp + ((m + 32) << 1));
        v2f b3 = *(const v2f*)(wp + ((m + 48) << 1));
        acc0 = __builtin_amdgcn_wmma_f32_16x16x4_f32(false, a, false, b0, (short)0, acc0, false, false);
        acc1 = __builtin_amdgcn_wmma_f32_16x16x4_f32(false, a, false, b1, (short)0, acc1, false, false);
        acc2 = __builtin_amdgcn_wmma_f32_16x16x4_f32(false, a, false, b2, (short)0, acc2, false, false);
        acc3 = __builtin_amdgcn_wmma_f32_16x16x4_f32(false, a, false, b3, (short)0, acc3, false, false);
    }

    // C/D layout: VGPR r -> M = r + half*8, N = lane&15 (+ tile*16)
    #pragma unroll
    for (int r = 0; r < 8; ++r) {
        const int rowOut = rowBase + r + half * 8;
        if (rowOut < nRows) {
            float* xo = X + (size_t)rowOut * H_F + m;
            xo[0]  = acc0[r];
            xo[16] = acc1[r];
            xo[32] = acc2[r];
            xo[48] = acc3[r];
        }
    }
}

// ---------------- edge propagation: h[dst] += x[src] * dinv[src]*dinv[dst] ----------------
// 16 threads per edge, 4 feats (float4) per thread.

__global__ void k_prop(const float* __restrict__ x, const int* __restrict__ src,
                       const int* __restrict__ dst, const float* __restrict__ dnv,
                       float* __restrict__ h, int nE) {
    int gid = blockIdx.x * blockDim.x + threadIdx.x;
    int e = gid >> 4;
    if (e >= nE) return;
    int f = (gid & 15) << 2;
    int s = src[e];
    int d = dst[e];
    float w = dnv[s] * dnv[d];
    const float4 xv = *(const float4*)(x + (size_t)s * H_F + f);
    float* hp = h + (size_t)d * H_F + f;
    unsafeAtomicAdd(hp + 0, xv.x * w);  // global_atomic_add_f32
    unsafeAtomicAdd(hp + 1, xv.y * w);
    unsafeAtomicAdd(hp + 2, xv.z * w);
    unsafeAtomicAdd(hp + 3, xv.w * w);
}

// self-loop term of GCNConv step: h[v] += x[v] * dinv1[v]^2   (non-atomic, runs after k_prop)
__global__ void k_selfloop(const float* __restrict__ x, const float* __restrict__ dinv1,
                           float* __restrict__ h, int n4) {
    int i = blockIdx.x * blockDim.x + threadIdx.x;
    if (i >= n4) return;
    int v = i >> 4;  // (i*4)/64
    float w = dinv1[v];
    w *= w;
    const float4 xv = ((const float4*)x)[i];
    float4 hv = ((float4*)h)[i];
    hv.x += xv.x * w; hv.y += xv.y * w; hv.z += xv.z * w; hv.w += xv.w * w;
    ((float4*)h)[i] = hv;
}

// out = softmax(alphas)[0]*h1 + softmax(alphas)[1]*h2
__global__ void k_combine(const float* __restrict__ h1, const float* __restrict__ h2,
                          const float* __restrict__ alphas, float* __restrict__ out, int n4) {
    int i = blockIdx.x * blockDim.x + threadIdx.x;
    if (i >= n4) return;
    float a0 = alphas[0], a1 = alphas[1];
    float mx = fmaxf(a0, a1);
    float e0 = __expf(a0 - mx), e1 = __expf(a1 - mx);
    float inv = 1.f / (e0 + e1);
    float w0 = e0 * inv, w1 = e1 * inv;
    const float4 u = ((const float4*)h1)[i];
    const float4 v = ((const float4*)h2)[i];
    float4 r;
    r.x = w0 * u.x + w1 * v.x;
    r.y = w0 * u.y + w1 * v.y;
    r.z = w0 * u.z + w1 * v.z;
    r.w = w0 * u.w + w1 * v.w;
    ((float4*)out)[i] = r;
}

// ---------------- launch ----------------

extern "C" void kernel_launch(void* const* d_in, const int* in_sizes, int n_in,
                              void* d_out, int out_size, void* d_ws, size_t ws_size,
                              hipStream_t stream) {
    const int* edge     = (const int*)d_in[0];   // [2, E] : first E = src, next E = dst
    const float* xin    = (const float*)d_in[1]; // [N, 128]
    const float* W1     = (const float*)d_in[2]; // [128, 64]
    const float* alphas = (const float*)d_in[3]; // [2]
    float* out          = (float*)d_out;         // [N, 64]

    const int E = N_EDGES, N = N_NODES;
    const int* src = edge;
    const int* dst = edge + E;

    float* ws    = (float*)d_ws;
    float* x     = ws;                      // [N*64]  (reused as h2 later)
    float* h1    = ws + (size_t)N * H_F;    // [N*64]
    int*   cnt   = (int*)(ws + 2 * (size_t)N * H_F);
    float* dinv1 = ws + 2 * (size_t)N * H_F + N;
    float* dinv2 = ws + 2 * (size_t)N * H_F + 2 * N;
    float* h2    = x;                       // alias: x dead after k_selfloop

    const int NF4 = (N * H_F) / 4;          // float4 count of a [N,64] buffer

    // degrees -> normalization tables
    k_zero_i32<<<(N + 255) / 256, 256, 0, stream>>>(cnt, N);
    k_count_deg<<<(E + 255) / 256, 256, 0, stream>>>(dst, cnt, E);
    k_dinv<<<(N + 255) / 256, 256, 0, stream>>>(cnt, dinv1, dinv2, N);

    // x = in_feat @ W1  (WMMA f32)
    k_gemm_wmma<<<(N + 127) / 128, 256, 0, stream>>>(xin, W1, x, N);

    // step 1: GCNConv propagation (with self loops) -> h1
    k_zero_f4<<<(NF4 + 255) / 256, 256, 0, stream>>>((float4*)h1, NF4);
    {
        long long t = (long long)E * 16;
        k_prop<<<(unsigned)((t + 255) / 256), 256, 0, stream>>>(x, src, dst, dinv1, h1, E);
    }
    k_selfloop<<<(NF4 + 255) / 256, 256, 0, stream>>>(x, dinv1, h1, NF4);

    // step 2: LGConv propagation (no self loops) -> h2 (aliases x)
    k_zero_f4<<<(NF4 + 255) / 256, 256, 0, stream>>>((float4*)h2, NF4);
    {
        long long t = (long long)E * 16;
        k_prop<<<(unsigned)((t + 255) / 256), 256, 0, stream>>>(h1, src, dst, dinv2, h2, E);
    }

    // out = softmax(alphas)[0]*h1 + softmax(alphas)[1]*h2
    k_combine<<<(NF4 + 255) / 256, 256, 0, stream>>>(h1, h2, alphas, out, NF4);
}